// DGCNN_77790447665944
// MI455X (gfx1250) — compile-verified
//
#include <hip/hip_runtime.h>

typedef __attribute__((ext_vector_type(2))) float v2f;
typedef __attribute__((ext_vector_type(8))) float v8f;

#define NPTS 2048
#define NB   4
#define ROWS 8192   // NB*NPTS
#define KNN  20
#define NEGINF -3.4e38f

// ---------------------------------------------------------------------------
// (B,3,N) -> (B,N,3)
__global__ void transpose_x(const float* __restrict__ x, float* __restrict__ xt) {
    int t = blockIdx.x * blockDim.x + threadIdx.x;   // over ROWS
    if (t >= ROWS) return;
    int b = t >> 11, n = t & 2047;
    #pragma unroll
    for (int c = 0; c < 3; ++c)
        xt[t * 3 + c] = x[((size_t)b * 3 + c) * NPTS + n];
}

// ---------------------------------------------------------------------------
// kNN: one wave (32 lanes) per query point, 4 waves per block.
// score(m) = 2*dot(xq,xm) - |xm|^2  (same ordering as -dist^2, const dropped)
__launch_bounds__(128)
__global__ void knn_kernel(const float* __restrict__ P, int ld, int coff,
                           int* __restrict__ idx, int C) {
    const int wave = threadIdx.x >> 5;
    const int lane = threadIdx.x & 31;
    const int r = blockIdx.x * 4 + wave;      // [0, ROWS)
    const int b = r >> 11;
    const int n = r & 2047;
    const float* Pb = P + (size_t)b * NPTS * ld + coff;
    const float* xq = Pb + (size_t)n * ld;

    __shared__ float s_score[4][NPTS];        // 32 KB
    float* sc = s_score[wave];

    for (int m = lane; m < NPTS; m += 32) {
        const float* xm = Pb + (size_t)m * ld;
        float dot = 0.f, sq = 0.f;
        for (int c = 0; c < C; ++c) {
            float v = xm[c];
            dot = fmaf(xq[c], v, dot);
            sq  = fmaf(v, v, sq);
        }
        sc[m] = 2.f * dot - sq;
    }
    // 20 argmax rounds (ties -> smallest index, like jax.lax.top_k)
    for (int t = 0; t < KNN; ++t) {
        float best = NEGINF; int bi = 0x7fffffff;
        for (int m = lane; m < NPTS; m += 32) {
            float v = sc[m];
            if (v > best) { best = v; bi = m; }
        }
        #pragma unroll
        for (int off = 16; off; off >>= 1) {
            float ov = __shfl_down(best, off);
            int   oi = __shfl_down(bi, off);
            if (ov > best || (ov == best && oi < bi)) { best = ov; bi = oi; }
        }
        bi = __shfl(bi, 0);
        if (lane == 0) idx[r * KNN + t] = bi;
        sc[bi] = NEGINF;   // all lanes write same value
    }
}

// ---------------------------------------------------------------------------
// Y[r, o] = sum_i X[r*ld_x + xoff + i] * W[o*ld_w + woff + i]
// One wave per 64x16 output strip (4 M-tiles), fp32 WMMA, K stepped by 4.
// B fragment (weights) loaded once per K-step, reused by 4 WMMAs.
// Software-pipelined: next K-step's 5 loads are issued before the current
// K-step's 4 WMMAs consume their fragments (staggered s_wait_loadcnt).
// rows % 64 == 0, O % 16 == 0, I % 4 == 0.
__launch_bounds__(256)
__global__ void wmma_gemm_xwT(const float* __restrict__ X, int ld_x, int xoff,
                              const float* __restrict__ W, int ld_w, int woff,
                              float* __restrict__ Y, int ld_y,
                              int rows, int O, int I) {
    const int wave = threadIdx.x >> 5;
    const int lane = threadIdx.x & 31;
    const int tiles_o = O >> 4;
    const int tile_id = blockIdx.x * 8 + wave;
    if (tile_id >= (rows >> 6) * tiles_o) return;   // wave-uniform exit
    const int m0 = (tile_id / tiles_o) << 6;        // 64-row strip
    const int o0 = (tile_id % tiles_o) << 4;

    const int l15 = lane & 15;
    const int hi  = lane >> 4;
    // A 16x4 layout: lane<16 holds (K,K+1) of row l15; lane>=16 holds (K+2,K+3)
    const float* arow = X + (size_t)(m0 + l15) * ld_x + xoff + hi * 2;
    const size_t astep = (size_t)16 * ld_x;
    // B 4x16 layout: B[kk][nn] = W[o0+nn][k+kk]
    const float* brow = W + (size_t)(o0 + l15) * ld_w + woff + hi * 2;

    v8f acc0 = {}, acc1 = {}, acc2 = {}, acc3 = {};

    // prologue: fragments for k = 0
    v2f bc  = *(const v2f*)(brow);
    v2f a0c = *(const v2f*)(arow);
    v2f a1c = *(const v2f*)(arow + astep);
    v2f a2c = *(const v2f*)(arow + 2 * astep);
    v2f a3c = *(const v2f*)(arow + 3 * astep);

    for (int k = 4; k < I; k += 4) {
        // issue next K-step's loads first (stay in flight during the WMMAs)
        v2f bn  = *(const v2f*)(brow + k);
        v2f a0n = *(const v2f*)(arow + k);
        v2f a1n = *(const v2f*)(arow + astep + k);
        v2f a2n = *(const v2f*)(arow + 2 * astep + k);
        v2f a3n = *(const v2f*)(arow + 3 * astep + k);
        acc0 = __builtin_amdgcn_wmma_f32_16x16x4_f32(false, a0c, false, bc, (short)0, acc0, false, false);
        acc1 = __builtin_amdgcn_wmma_f32_16x16x4_f32(false, a1c, false, bc, (short)0, acc1, false, false);
        acc2 = __builtin_amdgcn_wmma_f32_16x16x4_f32(false, a2c, false, bc, (short)0, acc2, false, false);
        acc3 = __builtin_amdgcn_wmma_f32_16x16x4_f32(false, a3c, false, bc, (short)0, acc3, false, false);
        bc = bn; a0c = a0n; a1c = a1n; a2c = a2n; a3c = a3n;
    }
    acc0 = __builtin_amdgcn_wmma_f32_16x16x4_f32(false, a0c, false, bc, (short)0, acc0, false, false);
    acc1 = __builtin_amdgcn_wmma_f32_16x16x4_f32(false, a1c, false, bc, (short)0, acc1, false, false);
    acc2 = __builtin_amdgcn_wmma_f32_16x16x4_f32(false, a2c, false, bc, (short)0, acc2, false, false);
    acc3 = __builtin_amdgcn_wmma_f32_16x16x4_f32(false, a3c, false, bc, (short)0, acc3, false, false);

    // C/D layout: reg v -> row m0 + 16*j + v + hi*8, col o0 + l15
    float* yrow = Y + (size_t)(m0 + hi * 8) * ld_y + o0 + l15;
    const size_t ystep = (size_t)16 * ld_y;
    #pragma unroll
    for (int v = 0; v < 8; ++v) {
        yrow[(size_t)v * ld_y]             = acc0[v];
        yrow[(size_t)v * ld_y + ystep]     = acc1[v];
        yrow[(size_t)v * ld_y + 2 * ystep] = acc2[v];
        yrow[(size_t)v * ld_y + 3 * ystep] = acc3[v];
    }
}

// Layer 1 GEMM (I = 3, too small for WMMA K-step of 4): O = 64
__global__ void gemm3(const float* __restrict__ X, const float* __restrict__ W,
                      int ldw, int woff, float* __restrict__ Y) {
    int t = blockIdx.x * blockDim.x + threadIdx.x;  // ROWS*64
    if (t >= ROWS * 64) return;
    int o = t & 63, r = t >> 6;
    const float* xp = X + (size_t)r * 3;
    const float* wp = W + (size_t)o * ldw + woff;
    Y[t] = fmaf(xp[0], wp[0], fmaf(xp[1], wp[1], xp[2] * wp[2]));
}

// ---------------------------------------------------------------------------
// For each (row, channel): h_k = Y1[gather] + Y2; accumulate BN stats over all
// 20 edges (atomics), store extremum (max if g>=0 else min, so BN+max commute).
__launch_bounds__(256)
__global__ void edge_agg(const float* __restrict__ Y1, const float* __restrict__ Y2,
                         const int* __restrict__ idx, const float* __restrict__ g,
                         float* __restrict__ E, float* __restrict__ stats, int O) {
    const int o    = threadIdx.x % O;
    const int nsub = threadIdx.x / O;
    const int npar = 256 / O;
    const int r0 = (blockIdx.x * npar + nsub) * 16;
    const bool useMax = (g[o] >= 0.f);
    float sum = 0.f, sq = 0.f;
    for (int s = 0; s < 16; ++s) {
        int r = r0 + s;
        if (r >= ROWS) break;
        int base = (r >> 11) << 11;                 // batch row base
        float y2 = Y2[(size_t)r * O + o];
        const int* ip = idx + r * KNN;
        float mx = NEGINF, mn = -NEGINF;
        #pragma unroll
        for (int kk = 0; kk < KNN; ++kk) {
            float h = Y1[(size_t)(base + ip[kk]) * O + o] + y2;
            sum += h; sq = fmaf(h, h, sq);
            mx = fmaxf(mx, h); mn = fminf(mn, h);
        }
        E[(size_t)r * O + o] = useMax ? mx : mn;
    }
    atomicAdd(&stats[o], sum);
    atomicAdd(&stats[O + o], sq);
}

__global__ void bn_final(const float* __restrict__ stats, const float* __restrict__ g,
                         const float* __restrict__ bb, float* __restrict__ scsh,
                         int O, float invCount) {
    int o = blockIdx.x * blockDim.x + threadIdx.x;
    if (o >= O) return;
    float mean = stats[o] * invCount;
    float var  = stats[O + o] * invCount - mean * mean;
    float sc = g[o] * rsqrtf(var + 1e-5f);
    scsh[o] = sc;
    scsh[O + o] = bb[o] - mean * sc;
}

__global__ void bn_apply(const float* __restrict__ E, const float* __restrict__ scsh,
                         float* __restrict__ out, int O, int ld_out, int ooff, int total) {
    int t = blockIdx.x * blockDim.x + threadIdx.x;
    if (t >= total) return;
    int o = t % O, r = t / O;
    float v = fmaf(E[t], scsh[o], scsh[O + o]);
    out[(size_t)r * ld_out + ooff + o] = (v >= 0.f) ? v : 0.2f * v;
}

// Layer-5 BN stats: one block per channel (deterministic tree reduce)
__launch_bounds__(256)
__global__ void col_stats(const float* __restrict__ H, float* __restrict__ stats, int O) {
    int o = blockIdx.x;
    float sum = 0.f, sq = 0.f;
    for (int r = threadIdx.x; r < ROWS; r += 256) {
        float v = H[(size_t)r * O + o];
        sum += v; sq = fmaf(v, v, sq);
    }
    __shared__ float ssum[256], ssq[256];
    ssum[threadIdx.x] = sum; ssq[threadIdx.x] = sq;
    __syncthreads();
    for (int s = 128; s; s >>= 1) {
        if (threadIdx.x < s) {
            ssum[threadIdx.x] += ssum[threadIdx.x + s];
            ssq[threadIdx.x]  += ssq[threadIdx.x + s];
        }
        __syncthreads();
    }
    if (threadIdx.x == 0) { stats[o] = ssum[0]; stats[O + o] = ssq[0]; }
}

// BN + lrelu + transpose to (B, 512, N); coalesced writes
__global__ void bn_apply_tr(const float* __restrict__ H, const float* __restrict__ scsh,
                            float* __restrict__ out) {
    int t = blockIdx.x * blockDim.x + threadIdx.x;  // ROWS*512
    if (t >= ROWS * 512) return;
    int n = t & 2047;
    int o = (t >> 11) & 511;
    int b = t >> 20;
    float v = fmaf(H[((size_t)(b << 11 | n)) * 512 + o], scsh[o], scsh[512 + o]);
    out[t] = (v >= 0.f) ? v : 0.2f * v;
}

// ---------------------------------------------------------------------------
extern "C" void kernel_launch(void* const* d_in, const int* in_sizes, int n_in,
                              void* d_out, int out_size, void* d_ws, size_t ws_size,
                              hipStream_t stream) {
    (void)in_sizes; (void)n_in; (void)out_size; (void)ws_size;
    const float* x  = (const float*)d_in[0];
    const float* W1 = (const float*)d_in[1];
    const float* g1 = (const float*)d_in[2];
    const float* b1 = (const float*)d_in[3];
    const float* W2 = (const float*)d_in[4];
    const float* g2 = (const float*)d_in[5];
    const float* b2 = (const float*)d_in[6];
    const float* W3 = (const float*)d_in[7];
    const float* g3 = (const float*)d_in[8];
    const float* b3 = (const float*)d_in[9];
    const float* W4 = (const float*)d_in[10];
    const float* g4 = (const float*)d_in[11];
    const float* b4 = (const float*)d_in[12];
    const float* W5 = (const float*)d_in[13];
    const float* g5 = (const float*)d_in[14];
    const float* b5 = (const float*)d_in[15];
    float* out = (float*)d_out;

    float* wsf = (float*)d_ws;
    float* XT  = wsf;                              // 8192*3
    float* CAT = XT  + (size_t)ROWS * 3;           // 8192*512 (x1|x2|x3|x4)
    float* YA  = CAT + (size_t)ROWS * 512;         // 8192*512 (Y1 / H5)
    float* YB  = YA  + (size_t)ROWS * 512;         // 8192*256 (Y2)
    float* E   = YB  + (size_t)ROWS * 256;         // 8192*256 (extrema)
    float* ST  = E   + (size_t)ROWS * 256;         // 1024 (sum, sumsq)
    float* SCB = ST  + 1024;                       // 1024 (scale, shift)
    int*   IDX = (int*)(SCB + 1024);               // 8192*20

    transpose_x<<<(ROWS + 255) / 256, 256, 0, stream>>>(x, XT);

    struct LayerDesc { int ld, xoff, C, O, outoff; const float *W, *g, *b; };
    const LayerDesc L[4] = {
        {3,   0,   3,   64,  0,   W1, g1, b1},
        {512, 0,   64,  64,  64,  W2, g2, b2},
        {512, 64,  64,  128, 128, W3, g3, b3},
        {512, 128, 128, 256, 256, W4, g4, b4},
    };

    for (int l = 0; l < 4; ++l) {
        const float* Xin = (l == 0) ? XT : CAT;
        knn_kernel<<<ROWS / 4, 128, 0, stream>>>(Xin, L[l].ld, L[l].xoff, IDX, L[l].C);
        if (l == 0) {
            gemm3<<<(ROWS * 64 + 255) / 256, 256, 0, stream>>>(XT, W1, 6, 0, YA);
            gemm3<<<(ROWS * 64 + 255) / 256, 256, 0, stream>>>(XT, W1, 6, 3, YB);
        } else {
            int strips = (ROWS / 64) * (L[l].O / 16);
            wmma_gemm_xwT<<<(strips + 7) / 8, 256, 0, stream>>>(
                CAT, 512, L[l].xoff, L[l].W, 2 * L[l].C, 0,
                YA, L[l].O, ROWS, L[l].O, L[l].C);
            wmma_gemm_xwT<<<(strips + 7) / 8, 256, 0, stream>>>(
                CAT, 512, L[l].xoff, L[l].W, 2 * L[l].C, L[l].C,
                YB, L[l].O, ROWS, L[l].O, L[l].C);
        }
        hipMemsetAsync(ST, 0, 2 * L[l].O * sizeof(float), stream);
        int npar = 256 / L[l].O;
        int blocks = (ROWS + npar * 16 - 1) / (npar * 16);
        edge_agg<<<blocks, 256, 0, stream>>>(YA, YB, IDX, L[l].g, E, ST, L[l].O);
        bn_final<<<(L[l].O + 255) / 256, 256, 0, stream>>>(
            ST, L[l].g, L[l].b, SCB, L[l].O, 1.0f / (float)(ROWS * KNN));
        bn_apply<<<(ROWS * L[l].O + 255) / 256, 256, 0, stream>>>(
            E, SCB, CAT, L[l].O, 512, L[l].outoff, ROWS * L[l].O);
    }

    // Layer 5: H = CAT @ W5^T, BN over (B,N), lrelu, transpose out
    int strips5 = (ROWS / 64) * (512 / 16);
    wmma_gemm_xwT<<<(strips5 + 7) / 8, 256, 0, stream>>>(
        CAT, 512, 0, W5, 512, 0, YA, 512, ROWS, 512, 512);
    col_stats<<<512, 256, 0, stream>>>(YA, ST, 512);
    bn_final<<<2, 256, 0, stream>>>(ST, g5, b5, SCB, 512, 1.0f / (float)ROWS);
    bn_apply_tr<<<(ROWS * 512 + 255) / 256, 256, 0, stream>>>(YA, SCB, out);
}